// GCN_Net_52261162057813
// MI455X (gfx1250) — compile-verified
//
#include <hip/hip_runtime.h>
#include <math.h>

// ---------------------------------------------------------------------------
// GCN (SGConv + 2x GCNConv + linear + log_softmax) for MI455X / gfx1250.
// Sparse prop is L2-resident (h <= 25.6MB, edges 6.4MB << 192MB L2);
// dense GEMMs use fp32 WMMA (v_wmma_f32_16x16x4_f32) to match fp32 reference.
// All GEMM strides are compile-time so loads lower to imm-offset global loads.
// ---------------------------------------------------------------------------

#define NNODES 50000
#define NEDGES 800000
#define FIN    128
#define HID    96
#define CLS    40

typedef float v2f __attribute__((ext_vector_type(2)));
typedef float v8f __attribute__((ext_vector_type(8)));

// ---- degree / normalization ------------------------------------------------
__global__ __launch_bounds__(256) void k_init_deg(float* deg, int n) {
    int i = blockIdx.x * blockDim.x + threadIdx.x;
    if (i < n) deg[i] = 1.0f;                       // self-loop contribution
}

__global__ __launch_bounds__(256) void k_deg_scatter(const int* __restrict__ col,
                                                     float* deg, int e) {
    int i = blockIdx.x * blockDim.x + threadIdx.x;
    if (i < e) atomicAdd(&deg[col[i]], 1.0f);
}

__global__ __launch_bounds__(256) void k_dinv(float* deg, int n) {
    int i = blockIdx.x * blockDim.x + threadIdx.x;
    if (i < n) deg[i] = rsqrtf(fmaxf(deg[i], 1.0f));
}

// ---- prop(): self-loop term written as initial value (no memset needed) ----
template <int F>
__global__ __launch_bounds__(256) void k_selfloop(const float* __restrict__ h,
                                                  const float* __restrict__ dinv,
                                                  float* __restrict__ out, int n) {
    int idx = blockIdx.x * blockDim.x + threadIdx.x;
    if (idx < n * F) {
        int i = idx / F;                            // F compile-time -> mul/shift
        float d = dinv[i];
        out[idx] = h[idx] * d * d;
    }
}

// one wave per edge; lanes stride the feature dimension (contiguous per row)
template <int F>
__global__ __launch_bounds__(256) void k_edge_prop(const float* __restrict__ h,
                                                   const int* __restrict__ row,
                                                   const int* __restrict__ colv,
                                                   const float* __restrict__ dinv,
                                                   float* __restrict__ out, int e) {
    int w    = (blockIdx.x * blockDim.x + threadIdx.x) >> 5;
    int lane = threadIdx.x & 31;
    if (w >= e) return;
    int r = row[w];
    int c = colv[w];
    float nw = dinv[r] * dinv[c];
    const float* hs = h   + (size_t)r * F + lane;
    float*       od = out + (size_t)c * F + lane;
#pragma unroll
    for (int j = 0; j < F / 32; ++j)
        atomicAdd(&od[32 * j], hs[32 * j] * nw);
}

// ---- fp32 WMMA GEMM: one wave computes a 16x32 C tile (two 16x16 WMMAs) ----
// A: M x K row-major (ld = K), W: K x NCOLS row-major, C: M x NCOLS.
// MODE: 0 = A@W + bias, 1 = relu(A@W + bias), 2 = A@W (no bias)
// ISA 7.12.2 (32-bit A 16x4): lane = 16*(k>=2) + m, vgpr = k&1; B mirrored
// (lane = 16*(k>=2) + n). C: vgpr v -> m = v + 8*(lane>=16), n = lane&15.
// For partial N tiles: a B element of column n only contributes to C column n,
// so out-of-range lanes just clamp their load address (finite garbage lands in
// columns we never store) -> no predication in the K loop.
template <int K, int NCOLS, int MODE>
__global__ __launch_bounds__(32) void k_gemm_wmma(const float* __restrict__ A,
                                                  const float* __restrict__ W,
                                                  const float* __restrict__ bias,
                                                  float* __restrict__ C) {
    const int mbase = blockIdx.x * 16;
    const int nbase = blockIdx.y * 32;              // two 16-wide N subtiles
    const int lane  = threadIdx.x;
    const int half  = lane >> 4;                    // K pair {0,1} vs {2,3}
    const int lm    = lane & 15;

    const int n0 = nbase + lm;
    const int n1 = nbase + 16 + lm;
    const int c0 = (n0 < NCOLS) ? n0 : (NCOLS - 1); // clamp: memory-safe
    const int c1 = (n1 < NCOLS) ? n1 : (NCOLS - 1);

    const float* __restrict__ Arow = A + (size_t)(mbase + lm) * K + 2 * half;
    const float* __restrict__ Wb0  = W + (size_t)(2 * half) * NCOLS + c0;
    const float* __restrict__ Wb1  = W + (size_t)(2 * half) * NCOLS + c1;

    v8f acc0 = {};
    v8f acc1 = {};
#pragma unroll
    for (int kk = 0; kk < K; kk += 4) {
        v2f a, b0, b1;
        a.x  = Arow[kk + 0];                        // imm-offset b64 load
        a.y  = Arow[kk + 1];
        b0.x = Wb0[(kk + 0) * NCOLS];               // imm-offset b32 loads
        b0.y = Wb0[(kk + 1) * NCOLS];
        b1.x = Wb1[(kk + 0) * NCOLS];
        b1.y = Wb1[(kk + 1) * NCOLS];
        acc0 = __builtin_amdgcn_wmma_f32_16x16x4_f32(false, a, false, b0,
                                                     (short)0, acc0, false, false);
        acc1 = __builtin_amdgcn_wmma_f32_16x16x4_f32(false, a, false, b1,
                                                     (short)0, acc1, false, false);
    }

    float bv0 = 0.0f, bv1 = 0.0f;
    if (MODE != 2) { bv0 = bias[c0]; bv1 = bias[c1]; }

#pragma unroll
    for (int v = 0; v < 8; ++v) {
        const int m = mbase + v + 8 * half;
        float v0 = acc0[v] + bv0;
        float v1 = acc1[v] + bv1;
        if (MODE == 1) { v0 = fmaxf(v0, 0.0f); v1 = fmaxf(v1, 0.0f); }
        if (n0 < NCOLS) C[(size_t)m * NCOLS + n0] = v0;   // folds away if full
        if (n1 < NCOLS) C[(size_t)m * NCOLS + n1] = v1;
    }
}

// ---- residual + bias + relu: H = relu(H + M + bg) --------------------------
__global__ __launch_bounds__(256) void k_resid_relu(float* __restrict__ H,
                                                    const float* __restrict__ M,
                                                    const float* __restrict__ bg,
                                                    int n) {
    int idx = blockIdx.x * blockDim.x + threadIdx.x;
    if (idx < n * HID) {
        float v = H[idx] + M[idx] + bg[idx % HID];
        H[idx] = fmaxf(v, 0.0f);
    }
}

// ---- in-place row-wise log_softmax over CLS columns ------------------------
__global__ __launch_bounds__(256) void k_log_softmax(float* __restrict__ L, int n) {
    int r = blockIdx.x * blockDim.x + threadIdx.x;
    if (r >= n) return;
    float* p = L + (size_t)r * CLS;
    float mx = -INFINITY;
#pragma unroll
    for (int c = 0; c < CLS; ++c) mx = fmaxf(mx, p[c]);
    float s = 0.0f;
#pragma unroll
    for (int c = 0; c < CLS; ++c) s += expf(p[c] - mx);
    float lse = logf(s) + mx;
#pragma unroll
    for (int c = 0; c < CLS; ++c) p[c] = p[c] - lse;
}

// ---------------------------------------------------------------------------
extern "C" void kernel_launch(void* const* d_in, const int* in_sizes, int n_in,
                              void* d_out, int out_size, void* d_ws, size_t ws_size,
                              hipStream_t stream) {
    const float* x  = (const float*)d_in[0];
    const int*   ei = (const int*)d_in[1];   // (2, E): row = ei[0:E), col = ei[E:2E)
    const float* W1 = (const float*)d_in[2];
    const float* b1 = (const float*)d_in[3];
    const float* Wg = (const float*)d_in[4];
    const float* bg = (const float*)d_in[5];
    const float* Wl = (const float*)d_in[6];
    const float* bl = (const float*)d_in[7];
    float* out = (float*)d_out;

    // workspace layout (floats): dinv[N] | B0[N*128] | B1=H[N*96] | B2=M[N*96]
    size_t need = (size_t)NNODES * (1 + FIN + 2 * HID) * sizeof(float);
    if (ws_size < need) return;
    float* dinv = (float*)d_ws;
    float* B0   = dinv + NNODES;
    float* B1   = B0 + (size_t)NNODES * FIN;
    float* B2   = B1 + (size_t)NNODES * HID;

    const int* row = ei;
    const int* col = ei + NEDGES;

    const int tpb = 256;

    // 1) symmetric GCN normalization: dinv = rsqrt(deg), deg includes self-loops
    k_init_deg   <<<(NNODES + tpb - 1) / tpb, tpb, 0, stream>>>(dinv, NNODES);
    k_deg_scatter<<<(NEDGES + tpb - 1) / tpb, tpb, 0, stream>>>(col, dinv, NEDGES);
    k_dinv       <<<(NNODES + tpb - 1) / tpb, tpb, 0, stream>>>(dinv, NNODES);

    // 2) prop(x) -> B0   (N x 128)
    k_selfloop<FIN> <<<((size_t)NNODES * FIN + tpb - 1) / tpb, tpb, 0, stream>>>(x, dinv, B0, NNODES);
    k_edge_prop<FIN><<<(NEDGES * 32 + tpb - 1) / tpb, tpb, 0, stream>>>(x, row, col, dinv, B0, NEDGES);

    // 3) H = relu(prop(x) @ W1 + b1)   (N x 96)
    k_gemm_wmma<FIN, HID, 1><<<dim3(NNODES / 16, HID / 32), 32, 0, stream>>>(B0, W1, b1, B1);

    // 4) two GCNConv layers with residual: H = relu(H + prop(H@Wg[i]) + bg[i])
    for (int i = 0; i < 2; ++i) {
        k_gemm_wmma<HID, HID, 2><<<dim3(NNODES / 16, HID / 32), 32, 0, stream>>>(
            B1, Wg + (size_t)i * HID * HID, nullptr, B0);
        k_selfloop<HID> <<<((size_t)NNODES * HID + tpb - 1) / tpb, tpb, 0, stream>>>(B0, dinv, B2, NNODES);
        k_edge_prop<HID><<<(NEDGES * 32 + tpb - 1) / tpb, tpb, 0, stream>>>(B0, row, col, dinv, B2, NEDGES);
        k_resid_relu<<<((size_t)NNODES * HID + tpb - 1) / tpb, tpb, 0, stream>>>(B1, B2, bg + i * HID, NNODES);
    }

    // 5) logits = H @ Wl + bl -> d_out, then in-place log_softmax
    k_gemm_wmma<HID, CLS, 0><<<dim3(NNODES / 16, (CLS + 31) / 32), 32, 0, stream>>>(B1, Wl, bl, out);
    k_log_softmax<<<(NNODES + tpb - 1) / tpb, tpb, 0, stream>>>(out, NNODES);
}